// HashEmbedding_80530636800412
// MI455X (gfx1250) — compile-verified
//
#include <hip/hip_runtime.h>
#include <stdint.h>
#include <stddef.h>

// ---------------------------------------------------------------------------
// Multiresolution hash encoding (Instant-NGP style), 2D, F=2, 16 levels.
//   out[p, 2*l + f] = bilerp of 4 hashed table entries, per level.
// Levels 0..4 (tables <= 26.9KB) are staged into LDS with the CDNA5 Tensor
// Data Mover (tensor_load_to_lds) and gathered via ds_load_b64.
// Levels 5..15 gather from global memory (all tables are L2-resident; output
// uses non-temporal stores so the 256MB stream does not evict them).
// ---------------------------------------------------------------------------

#define N_LEVELS 16
#define N_LDS_LEVELS 5
#define P1C 2654435761ull
#define P2C 805459861ull

// res_i = floor(16 * 2^(7*i/15)); table = min(res^2, 2^19)
static constexpr uint32_t kRes[N_LEVELS] = {
    16, 22, 30, 42, 58, 80, 111, 153, 212, 294, 406, 561, 776, 1072, 1482, 2048};
static constexpr uint32_t kTs[N_LEVELS] = {
    256, 484, 900, 1764, 3364, 6400, 12321, 23409, 44944, 86436,
    164836, 314721, 524288, 524288, 524288, 524288};

#define MAX_LDS_TS 3364  // table size of level 4 (58^2), 26,912 bytes as float2

struct LevelParam {
  const float2* tab;
  uint64_t magic;   // floor(2^64 / ts) + 1  (exact div for v < 2^43, ts < 2^20)
  uint32_t ts;
  uint32_t res;
  float resf;
  uint32_t pad;
};
struct AllParams { LevelParam lv[N_LEVELS]; };

typedef unsigned int v4u __attribute__((ext_vector_type(4)));
typedef int          v8i __attribute__((ext_vector_type(8)));
typedef int          v4i __attribute__((ext_vector_type(4)));
typedef float        v2f __attribute__((ext_vector_type(2)));

// hash = (cx*P1 ^ cy*P2) mod ts, exact via magic multiply (no 64-bit divide).
__device__ __forceinline__ uint32_t hash_mod(uint32_t cx, uint32_t cy,
                                             uint64_t magic, uint32_t ts) {
  uint64_t v = ((uint64_t)cx * P1C) ^ ((uint64_t)cy * P2C);
  uint64_t q = __umul64hi(v, magic);
  return (uint32_t)(v - q * (uint64_t)ts);
}

__device__ __forceinline__ void corners(float xin, float resf, int res,
                                        uint32_t& c0, uint32_t& c1, float& w) {
  float xn = (xin + 1.0f) * 0.5f;            // (x - BOX_MIN)/(BOX_MAX-BOX_MIN)
  xn = fminf(fmaxf(xn, 0.0f), 1.0f);
  float xs = xn * resf;
  float xf = floorf(xs);
  w = xs - xf;
  int i0 = (int)xf;                          // in [0, res]
  int i1 = i0 + 1;
  c0 = (uint32_t)min(i0, res - 1);
  c1 = (uint32_t)min(i1, res - 1);
}

__device__ __forceinline__ float2 bilerp(float2 v00, float2 v10, float2 v01,
                                         float2 v11, float wx, float wy) {
  float v0x = v00.x + wx * (v10.x - v00.x);
  float v0y = v00.y + wx * (v10.y - v00.y);
  float v1x = v01.x + wx * (v11.x - v01.x);
  float v1y = v01.y + wx * (v11.y - v01.y);
  float2 r;
  r.x = v0x + wy * (v1x - v0x);
  r.y = v0y + wy * (v1y - v0y);
  return r;
}

__device__ __forceinline__ void store_nt(float* out, size_t p, uint32_t l, float2 r) {
  v2f rv; rv.x = r.x; rv.y = r.y;
  __builtin_nontemporal_store(rv, (v2f*)(out + p * (2 * N_LEVELS) + 2 * l));
}

// ---------------------------------------------------------------------------
// Kernel A: levels 0..4. blockIdx.y = level (uniform). Table staged into LDS
// by the Tensor Data Mover, then grid-stride over all points.
// ---------------------------------------------------------------------------
__global__ void __launch_bounds__(256)
hash_enc_lds(const float2* __restrict__ x, float* __restrict__ out,
             AllParams P, int npts) {
  const uint32_t l = blockIdx.y;
  const LevelParam lp = P.lv[l];
  __shared__ float2 smem[MAX_LDS_TS];

  if (threadIdx.x < 32) {  // one wave issues the DMA (TDM ignores EXEC)
    const uint64_t ga = (uint64_t)(uintptr_t)lp.tab;
    const uint32_t lds_base = (uint32_t)(uintptr_t)&smem[0];  // LDS byte offset
    const uint32_t ts = lp.ts;
    // D# group 0: count=1 (valid), lds_addr, global_addr[56:0], type=2
    v4u g0 = { 1u,
               lds_base,
               (uint32_t)ga,
               ((uint32_t)(ga >> 32) & 0x01FFFFFFu) | 0x80000000u };
    // D# group 1: data_size=3 (8B elems), tensor_dim0=ts, tensor_dim1=1,
    //             tile_dim0=ts (1-D tile), tensor_dim0_stride=ts
    v8i g1 = { (int)(3u << 16),
               (int)((ts & 0xFFFFu) << 16),        // tensor_dim0[15:0] @bits63:48
               (int)((ts >> 16) | (1u << 16)),     // tensor_dim0[31:16], dim1=1
               (int)(ts << 16),                    // tile_dim0 @bits127:112
               0,                                  // tile_dim1=0, tile_dim2=0
               (int)ts,                            // tensor_dim0_stride[31:0]
               0, 0 };
    v4i g2 = { 1, 1, 0, 0 };                       // tensor_dim2=1, tensor_dim3=1
    v4i g3 = { 0, (int)(1u << 16), 0, 0 };         // tensor_dim4=1
    v8i g4 = { 0, 0, 0, 0, 0, 0, 0, 0 };           // trailing group (unused)
    __builtin_amdgcn_tensor_load_to_lds(g0, g1, g2, g3, g4, 0);
    __builtin_amdgcn_s_wait_tensorcnt(0);
  }
  __syncthreads();

  const int stride = gridDim.x * blockDim.x;
  const int res = (int)lp.res;
  for (int p = blockIdx.x * blockDim.x + threadIdx.x; p < npts; p += stride) {
    float2 xv = x[p];
    uint32_t x0, x1, y0, y1; float wx, wy;
    corners(xv.x, lp.resf, res, x0, x1, wx);
    corners(xv.y, lp.resf, res, y0, y1, wy);
    float2 v00 = smem[hash_mod(x0, y0, lp.magic, lp.ts)];
    float2 v10 = smem[hash_mod(x1, y0, lp.magic, lp.ts)];
    float2 v01 = smem[hash_mod(x0, y1, lp.magic, lp.ts)];
    float2 v11 = smem[hash_mod(x1, y1, lp.magic, lp.ts)];
    store_nt(out, (size_t)p, l, bilerp(v00, v10, v01, v11, wx, wy));
  }
}

// ---------------------------------------------------------------------------
// Kernel B: levels 5..15. blockIdx.y + 5 = level (uniform). L2-resident
// global gathers; one (point, level) per thread.
// ---------------------------------------------------------------------------
__global__ void __launch_bounds__(256)
hash_enc_global(const float2* __restrict__ x, float* __restrict__ out,
                AllParams P, int npts) {
  const uint32_t l = N_LDS_LEVELS + blockIdx.y;
  const int p = blockIdx.x * blockDim.x + threadIdx.x;
  if (p >= npts) return;
  const LevelParam lp = P.lv[l];

  float2 xv = x[p];
  uint32_t x0, x1, y0, y1; float wx, wy;
  corners(xv.x, lp.resf, (int)lp.res, x0, x1, wx);
  corners(xv.y, lp.resf, (int)lp.res, y0, y1, wy);
  const float2* __restrict__ t = lp.tab;
  float2 v00 = t[hash_mod(x0, y0, lp.magic, lp.ts)];
  float2 v10 = t[hash_mod(x1, y0, lp.magic, lp.ts)];
  float2 v01 = t[hash_mod(x0, y1, lp.magic, lp.ts)];
  float2 v11 = t[hash_mod(x1, y1, lp.magic, lp.ts)];
  store_nt(out, (size_t)p, l, bilerp(v00, v10, v01, v11, wx, wy));
}

// ---------------------------------------------------------------------------
extern "C" void kernel_launch(void* const* d_in, const int* in_sizes, int n_in,
                              void* d_out, int out_size, void* d_ws, size_t ws_size,
                              hipStream_t stream) {
  (void)d_ws; (void)ws_size; (void)out_size; (void)n_in;
  const float2* x = (const float2*)d_in[0];
  float* out = (float*)d_out;
  const int npts = in_sizes[0] / 2;

  AllParams P;
  for (int i = 0; i < N_LEVELS; ++i) {
    P.lv[i].tab   = (const float2*)d_in[1 + i];
    P.lv[i].ts    = kTs[i];
    P.lv[i].res   = kRes[i];
    P.lv[i].resf  = (float)kRes[i];
    P.lv[i].magic = ~0ull / (uint64_t)kTs[i] + 1ull;  // floor(2^64/ts)+1
    P.lv[i].pad   = 0;
  }

  // Kernel A: persistent blocks amortize the LDS staging DMA.
  dim3 blkA(256), grdA(512, N_LDS_LEVELS);
  hipLaunchKernelGGL(hash_enc_lds, grdA, blkA, 0, stream, x, out, P, npts);

  // Kernel B: one (point, level) per thread.
  dim3 blkB(256), grdB((unsigned)((npts + 255) / 256), N_LEVELS - N_LDS_LEVELS);
  hipLaunchKernelGGL(hash_enc_global, grdB, blkB, 0, stream, x, out, P, npts);
}